// Fuse_SSM_51496657879068
// MI455X (gfx1250) — compile-verified
//
#include <hip/hip_runtime.h>
#include <hip/hip_bf16.h>

typedef float v2f __attribute__((ext_vector_type(2)));
typedef float v8f __attribute__((ext_vector_type(8)));
typedef int   v4i __attribute__((ext_vector_type(4)));

constexpr int B_SZ    = 2;
constexpr int L_SZ    = 1024;
constexpr int D_MODEL = 1024;
constexpr int D_INNER = 2048;
constexpr int DSTATE  = 16;
constexpr int DT_RANK = 64;
constexpr int NCOL    = B_SZ * L_SZ;            // 2048 (columns = b*L + l)

// ---------------------------------------------------------------------------
// gfx1250 async global->LDS copies (ASYNCcnt), guarded so we fall back to the
// plain load/ds_store path if this toolchain doesn't declare the builtins.
// Probe-learned signatures: (AS1 int* gsrc, AS3 int* ldst, int off, int cpol),
// b128 variant takes int4*.
// ---------------------------------------------------------------------------
#if __has_builtin(__builtin_amdgcn_global_load_async_to_lds_b32) && \
    __has_builtin(__builtin_amdgcn_global_load_async_to_lds_b128) && \
    __has_builtin(__builtin_amdgcn_s_wait_asynccnt)
#define HAS_ASYNC_LDS 1
#else
#define HAS_ASYNC_LDS 0
#endif

#if HAS_ASYNC_LDS
__device__ __forceinline__ void async_copy_b32(const float* g, float* l) {
  float* gg = const_cast<float*>(g);
  __builtin_amdgcn_global_load_async_to_lds_b32(
      (__attribute__((address_space(1))) int*)gg,
      (__attribute__((address_space(3))) int*)l, 0, 0);
}
__device__ __forceinline__ void async_copy_b128(const float* g, float* l) {
  float* gg = const_cast<float*>(g);
  __builtin_amdgcn_global_load_async_to_lds_b128(
      (__attribute__((address_space(1))) v4i*)gg,
      (__attribute__((address_space(3))) v4i*)l, 0, 0);
}
#endif

// ---------------------------------------------------------------------------
// f32 WMMA helpers (V_WMMA_F32_16X16X4_F32, wave32)
// A (16x4): lane holds row M=lane%16, K = {0,1} (lanes 0-15) or {2,3} (16-31)
// B (4x16): lane holds col N=lane%16, K = {0,1} / {2,3} split the same way
// C/D (16x16): VGPR v holds M = v + 8*(lane>=16), N = lane%16
// ---------------------------------------------------------------------------
__device__ __forceinline__ v8f wmma_f32(v2f a, v2f b, v8f c) {
  return __builtin_amdgcn_wmma_f32_16x16x4_f32(false, a, false, b,
                                               (short)0, c, false, false);
}

__device__ __forceinline__ v2f load_a_frag(const float* __restrict__ A, int lda,
                                           int r0, int k0, int lane) {
  int row = r0 + (lane & 15);
  int kb  = k0 + ((lane >> 4) << 1);   // 0 or 2
  const float* p = A + row * lda + kb;
  v2f a; a.x = p[0]; a.y = p[1];
  return a;
}

// ---------------------------------------------------------------------------
// K1: causal depthwise conv1d(k=4) + bias + SiLU, with (b,l,d)->(b,d,l)
// transpose through LDS.  grid (D_INNER/64, L/64, B), block (64,4).
// ---------------------------------------------------------------------------
__global__ void conv_silu_kernel(const float* __restrict__ hs,
                                 const float* __restrict__ cw,
                                 const float* __restrict__ cb,
                                 float* __restrict__ x_out) {
  __shared__ float tile[64][69];   // [d][l + 3 halo], odd stride -> conflict-free
  const int d0 = blockIdx.x * 64;
  const int l0 = blockIdx.y * 64;
  const int b  = blockIdx.z;
  const int tx = threadIdx.x;      // 0..63
  const int ty = threadIdx.y;      // 0..3

  for (int r = ty; r < 67; r += 4) {        // r=0 -> l0-3
    int l = l0 + r - 3;
    float v = (l >= 0) ? hs[(b * L_SZ + l) * D_INNER + d0 + tx] : 0.0f;
    tile[tx][r] = v;
  }
  __syncthreads();

  // compute: tx indexes l (coalesced stores), loop over d
  for (int d = ty; d < 64; d += 4) {
    float acc = 0.0f;
#pragma unroll
    for (int k = 0; k < 4; ++k)
      acc += cw[(d0 + d) * 4 + k] * tile[d][tx + k];
    acc += cb[d0 + d];
    float s = acc / (1.0f + __expf(-acc));   // SiLU
    x_out[((long)b * D_INNER + (d0 + d)) * L_SZ + l0 + tx] = s;
  }
}

// ---------------------------------------------------------------------------
// K2: x_dbl[r, n] = sum_d Wp[r,d] * x[b(n), d, l(n)]    (M=96, K=2048, N=2048)
// one wave per 16x64 output strip; grid(24), block(256) = 192 waves total.
// ---------------------------------------------------------------------------
__global__ void xproj_gemm_kernel(const float* __restrict__ Wp,
                                  const float* __restrict__ x,
                                  float* __restrict__ xdbl) {
  const int wid  = blockIdx.x * 8 + (threadIdx.x >> 5);
  const int lane = threadIdx.x & 31;
  const int rt = wid % 6;          // 6 row tiles (96/16)
  const int ng = wid / 6;          // 32 column groups of 64
  const int r0 = rt * 16;
  const int n0 = ng * 64;
  const int bq = n0 >> 10, lq = n0 & (L_SZ - 1);
  const float* xb = x + (long)bq * D_INNER * L_SZ + lq;

  const int nlane = lane & 15;
  const int khalf = (lane >> 4) << 1;
  const int hi8   = (lane >> 4) * 8;

  v8f acc[4] = {};
  for (int k0 = 0; k0 < D_INNER; k0 += 4) {
    v2f a = load_a_frag(Wp, D_INNER, r0, k0, lane);
#pragma unroll
    for (int j = 0; j < 4; ++j) {
      const float* bp = xb + (long)(k0 + khalf) * L_SZ + j * 16 + nlane;
      v2f bf; bf.x = bp[0]; bf.y = bp[L_SZ];
      acc[j] = wmma_f32(a, bf, acc[j]);
    }
  }
#pragma unroll
  for (int j = 0; j < 4; ++j) {
    int n = n0 + j * 16 + nlane;
#pragma unroll
    for (int v = 0; v < 8; ++v)
      xdbl[(r0 + hi8 + v) * NCOL + n] = acc[j][v];
  }
}

// ---------------------------------------------------------------------------
// K3: delta[b,d,l] = softplus( sum_r Wdt[d,r]*xdbl[r,n] + bias[d] )
// M=2048, K=64, N=2048.  grid(512), block(256) = 4096 waves.
// ---------------------------------------------------------------------------
__global__ void dtproj_softplus_kernel(const float* __restrict__ Wdt,
                                       const float* __restrict__ xdbl,
                                       const float* __restrict__ bias,
                                       float* __restrict__ delta) {
  const int wid  = blockIdx.x * 8 + (threadIdx.x >> 5);
  const int lane = threadIdx.x & 31;
  const int dt = wid & 127;        // 128 row tiles
  const int ng = wid >> 7;         // 32 column groups
  const int r0 = dt * 16;
  const int n0 = ng * 64;

  const int nlane = lane & 15;
  const int khalf = (lane >> 4) << 1;
  const int hi8   = (lane >> 4) * 8;

  v8f acc[4] = {};
#pragma unroll
  for (int k0 = 0; k0 < DT_RANK; k0 += 4) {
    v2f a = load_a_frag(Wdt, DT_RANK, r0, k0, lane);
#pragma unroll
    for (int j = 0; j < 4; ++j) {
      const float* bp = xdbl + (k0 + khalf) * NCOL + n0 + j * 16 + nlane;
      v2f bf; bf.x = bp[0]; bf.y = bp[NCOL];
      acc[j] = wmma_f32(a, bf, acc[j]);
    }
  }
  const int bq = n0 >> 10, lq = n0 & (L_SZ - 1);
  float* db = delta + (long)bq * D_INNER * L_SZ + lq;
#pragma unroll
  for (int j = 0; j < 4; ++j) {
#pragma unroll
    for (int v = 0; v < 8; ++v) {
      int m = r0 + hi8 + v;
      float val = acc[j][v] + bias[m];
      val = (val > 20.0f) ? val : logf(1.0f + __expf(val));
      db[(long)m * L_SZ + j * 16 + nlane] = val;
    }
  }
}

// ---------------------------------------------------------------------------
// K4: selective scan.  One thread per (b,d) channel, 16-wide state in regs,
// B/C staged in LDS per 16-step chunk via async global->LDS copies.
// Fuses y = (scan + D*u) * silu(z) and writes IN PLACE over x.
// grid(16), block(256): block = 256 consecutive channels of one batch.
// ---------------------------------------------------------------------------
__global__ void scan_kernel(const float* __restrict__ xdbl,
                            const float* __restrict__ delta,
                            float* __restrict__ xy,          // in: u, out: y
                            const float* __restrict__ z,
                            const float* __restrict__ A_log,
                            const float* __restrict__ Dp) {
  const int b = blockIdx.x >> 3;
  const int d = ((blockIdx.x & 7) << 8) + threadIdx.x;

  float a[DSTATE], h[DSTATE];
#pragma unroll
  for (int n = 0; n < DSTATE; ++n) {
    a[n] = -__expf(A_log[d * DSTATE + n]);
    h[n] = 0.0f;
  }
  const float Dd = Dp[d];

  const float* pB = xdbl + DT_RANK * NCOL + b * L_SZ;            // rows 64..79
  const float* pC = xdbl + (DT_RANK + DSTATE) * NCOL + b * L_SZ; // rows 80..95
  float*       px = xy    + ((long)b * D_INNER + d) * L_SZ;
  const float* pdl = delta + ((long)b * D_INNER + d) * L_SZ;
  const float* pz = z + (long)b * L_SZ * D_INNER + d;

  __shared__ float sB[16][17];
  __shared__ float sC[16][17];
  const int tj = threadIdx.x & 15, tn = threadIdx.x >> 4;

  for (int lc = 0; lc < L_SZ; lc += 16) {
    __syncthreads();
#if HAS_ASYNC_LDS
    async_copy_b32(pB + tn * NCOL + lc + tj, &sB[tj][tn]);
    async_copy_b32(pC + tn * NCOL + lc + tj, &sC[tj][tn]);
    __builtin_amdgcn_s_wait_asynccnt(0);
#else
    sB[tj][tn] = pB[tn * NCOL + lc + tj];
    sC[tj][tn] = pC[tn * NCOL + lc + tj];
#endif
    __syncthreads();
#pragma unroll
    for (int q = 0; q < 4; ++q) {
      float4 d4 = *(const float4*)(pdl + lc + q * 4);
      float4 u4 = *(const float4*)(px + lc + q * 4);
      float dl[4] = {d4.x, d4.y, d4.z, d4.w};
      float uu[4] = {u4.x, u4.y, u4.z, u4.w};
      float y4[4];
#pragma unroll
      for (int s = 0; s < 4; ++s) {
        const int jj = q * 4 + s;
        const float dlt = dl[s];
        const float du  = dlt * uu[s];
        float y = 0.0f;
#pragma unroll
        for (int n = 0; n < DSTATE; ++n) {
          float dA = __expf(dlt * a[n]);
          h[n] = dA * h[n] + du * sB[jj][n];
          y += h[n] * sC[jj][n];
        }
        float zv = pz[(long)(lc + jj) * D_INNER];
        float sz = zv / (1.0f + __expf(-zv));
        y4[s] = (y + Dd * uu[s]) * sz;
      }
      float4 o4 = make_float4(y4[0], y4[1], y4[2], y4[3]);
      *(float4*)(px + lc + q * 4) = o4;
    }
  }
}

// ---------------------------------------------------------------------------
// K5: out[n, o] = sum_d Wout[o,d] * y[b(n), d, l(n)]   (M=1024, K=2048, N=2048)
// Block = 8 waves, 128(M) x 64(N) per block; B panels staged in LDS via
// async b128 copies (stride 72 floats keeps half-wave K rows bank-disjoint
// and every row 16B-aligned).  grid(8, 32), block(256).
// ---------------------------------------------------------------------------
__global__ void outproj_gemm_kernel(const float* __restrict__ Wout,
                                    const float* __restrict__ y,
                                    float* __restrict__ out) {
  __shared__ __align__(16) float sY[32][72];
  const int w    = threadIdx.x >> 5;
  const int lane = threadIdx.x & 31;
  const int o0 = blockIdx.x * 128 + w * 16;
  const int n0 = blockIdx.y * 64;
  const int bq = n0 >> 10, lq = n0 & (L_SZ - 1);
  const float* yb = y + (long)bq * D_INNER * L_SZ + lq;

  const int nlane = lane & 15;
  const int khalf = (lane >> 4) << 1;
  const int hi8   = (lane >> 4) * 8;

#if HAS_ASYNC_LDS
  const int crow = threadIdx.x >> 4;        // 0..15
  const int ccol = (threadIdx.x & 15) * 4;  // float col, 16B granules
#else
  const int col   = threadIdx.x & 63;
  const int rbase = threadIdx.x >> 6;
#endif

  v8f acc[4] = {};
  for (int k0 = 0; k0 < D_INNER; k0 += 32) {
    __syncthreads();
#if HAS_ASYNC_LDS
    async_copy_b128(yb + (long)(k0 + crow) * L_SZ + ccol, &sY[crow][ccol]);
    async_copy_b128(yb + (long)(k0 + crow + 16) * L_SZ + ccol, &sY[crow + 16][ccol]);
    __builtin_amdgcn_s_wait_asynccnt(0);
#else
#pragma unroll
    for (int rr = rbase; rr < 32; rr += 4)
      sY[rr][col] = yb[(long)(k0 + rr) * L_SZ + col];
#endif
    __syncthreads();
#pragma unroll
    for (int kk = 0; kk < 32; kk += 4) {
      v2f a = load_a_frag(Wout, D_INNER, o0, k0 + kk, lane);
#pragma unroll
      for (int j = 0; j < 4; ++j) {
        v2f bf;
        bf.x = sY[kk + khalf][j * 16 + nlane];
        bf.y = sY[kk + khalf + 1][j * 16 + nlane];
        acc[j] = wmma_f32(a, bf, acc[j]);
      }
    }
  }
#pragma unroll
  for (int j = 0; j < 4; ++j) {
    int n = n0 + j * 16 + nlane;
#pragma unroll
    for (int v = 0; v < 8; ++v)
      out[(long)n * D_MODEL + o0 + hi8 + v] = acc[j][v];
  }
}

// ---------------------------------------------------------------------------
extern "C" void kernel_launch(void* const* d_in, const int* in_sizes, int n_in,
                              void* d_out, int out_size, void* d_ws, size_t ws_size,
                              hipStream_t stream) {
  const float* hs        = (const float*)d_in[0];  // (B, L, D_INNER)
  const float* z         = (const float*)d_in[1];  // (B, L, D_INNER)
  const float* conv_w    = (const float*)d_in[2];  // (D_INNER, 4)
  const float* conv_b    = (const float*)d_in[3];  // (D_INNER,)
  const float* x_proj_w  = (const float*)d_in[4];  // (96, D_INNER)
  const float* dt_proj_w = (const float*)d_in[5];  // (D_INNER, 64)
  const float* dt_proj_b = (const float*)d_in[6];  // (D_INNER,)
  const float* A_log     = (const float*)d_in[7];  // (D_INNER, 16)
  const float* Dp        = (const float*)d_in[8];  // (D_INNER,)
  const float* out_w     = (const float*)d_in[9];  // (D_MODEL, D_INNER)
  float* out = (float*)d_out;                       // (B, L, D_MODEL)

  float* ws    = (float*)d_ws;
  float* x     = ws;                                 // (B, D_INNER, L) -> later y
  float* delta = ws + (size_t)B_SZ * D_INNER * L_SZ; // (B, D_INNER, L)
  float* xdbl  = delta + (size_t)B_SZ * D_INNER * L_SZ; // (96, NCOL)

  conv_silu_kernel<<<dim3(D_INNER / 64, L_SZ / 64, B_SZ), dim3(64, 4), 0, stream>>>(
      hs, conv_w, conv_b, x);

  xproj_gemm_kernel<<<dim3(24), dim3(256), 0, stream>>>(x_proj_w, x, xdbl);

  dtproj_softplus_kernel<<<dim3(512), dim3(256), 0, stream>>>(
      dt_proj_w, xdbl, dt_proj_b, delta);

  scan_kernel<<<dim3(16), dim3(256), 0, stream>>>(xdbl, delta, x, z, A_log, Dp);

  outproj_gemm_kernel<<<dim3(8, 32), dim3(256), 0, stream>>>(out_w, x, out);
}